// RobustListLearner_52441550684463
// MI455X (gfx1250) — compile-verified
//
#include <hip/hip_runtime.h>
#include <cstdint>
#include <cstddef>

// ---------------------------------------------------------------------------
// RobustListLearner on MI455X (gfx1250, wave32)
//
// Output: dense [Cf*Cs, 24] f32 (~215 MB). Store-bandwidth bound (~9.6us at
// 23.3 TB/s). Strategy:
//   K1 prep   : lf[128x24], fid[276] pairs, sid[8128] pairs, rhs[8128]  (tiny)
//   K2 WMMA   : U = lf @ S0, V = lf @ S1 via V_WMMA_F32_16X16X4_F32
//               (S0/S1 are 0/1 selection matrices -> result is EXACT f32)
//   K3 writer : one wave per (f, 32 samples); 6 x fully-coalesced float4
//               non-temporal stores per lane; every output float written once.
// ---------------------------------------------------------------------------

typedef float v2f __attribute__((ext_vector_type(2)));
typedef float v4f __attribute__((ext_vector_type(4)));
typedef float v8f __attribute__((ext_vector_type(8)));

#define NSAMP   128
#define DFEAT   24
#define CS      8128            // C(128,2)
#define CF      276             // C(24,2)
#define CFPAD   288             // 18 tiles of 16
#define SCHUNKS 254             // CS / 32

// workspace byte offsets (total ~440 KB)
#define LF_OFF    0u            // float[128*24]
#define FC0_OFF   12288u        // int[276]
#define FC1_OFF   14336u        // int[276]
#define PQ_OFF    16384u        // int2[8128]
#define RHS_OFF   81920u        // float2[8128]
#define U_OFF     147456u       // float[128*288]
#define V_OFF     294912u       // float[128*288]

// ---------------------------------------------------------------- K1: prep
__global__ void rll_prep_kernel(const float* __restrict__ distr,
                                char* __restrict__ ws) {
    float*  lf  = (float*) (ws + LF_OFF);
    int*    fc0 = (int*)   (ws + FC0_OFF);
    int*    fc1 = (int*)   (ws + FC1_OFF);
    int2*   pq  = (int2*)  (ws + PQ_OFF);
    float2* rhs = (float2*)(ws + RHS_OFF);
    const int tid = threadIdx.x;

    // labeled features
    for (int idx = tid; idx < NSAMP * DFEAT; idx += blockDim.x) {
        int p = idx / DFEAT, d = idx - p * DFEAT;
        float lab = distr[p * 25] * 2.0f - 1.0f;
        lf[idx] = lab * distr[p * 25 + 1 + d];
    }
    // feature-pair combinations (lexicographic)
    for (int f = tid; f < CF; f += blockDim.x) {
        int rem = f, a = 0;
        while (rem >= (DFEAT - 1) - a) { rem -= (DFEAT - 1) - a; ++a; }
        fc0[f] = a;
        fc1[f] = a + 1 + rem;
    }
    // sample-pair combinations + rhs = labels(pair) - margin  (margin = 1)
    for (int s = tid; s < CS; s += blockDim.x) {
        int rem = s, p = 0;
        while (rem >= (NSAMP - 1) - p) { rem -= (NSAMP - 1) - p; ++p; }
        int q = p + 1 + rem;
        pq[s] = make_int2(p, q);
        float lp = distr[p * 25] * 2.0f - 1.0f;
        float lq = distr[q * 25] * 2.0f - 1.0f;
        rhs[s] = make_float2(lp - 1.0f, lq - 1.0f);
    }
}

// -------------------------------------------- K2: U,V via f32 WMMA (exact)
// U[p,f] = lf[p, fid[f,0]],  V[p,f] = lf[p, fid[f,1]]  computed as
// lf (128x24) @ S (24x288 one-hot) with V_WMMA_F32_16X16X4_F32, K in 6 steps.
// 144 waves = 8 M-tiles x 18 N-tiles; 18 blocks of 256 threads (8 waves).
__global__ void rll_uv_wmma_kernel(char* __restrict__ ws) {
    const float* lf  = (const float*)(ws + LF_OFF);
    const int*   fc0 = (const int*)  (ws + FC0_OFF);
    const int*   fc1 = (const int*)  (ws + FC1_OFF);
    float*       U   = (float*)      (ws + U_OFF);
    float*       V   = (float*)      (ws + V_OFF);

    const int wave  = blockIdx.x * (blockDim.x >> 5) + (threadIdx.x >> 5);
    const int lane  = threadIdx.x & 31;
    const int mtile = wave & 7;    // 0..7
    const int ntile = wave >> 3;   // 0..17

    // A layout (f32 16x4): lane l holds A[l&15, 2*(l>>4) + j] in vgpr j
    const int row   = mtile * 16 + (lane & 15);
    const int khalf = (lane >> 4) * 2;

    // B layout (f32 4x16): lane l holds B[2*(l>>4) + j, l&15] in vgpr j
    const int f      = ntile * 16 + (lane & 15);
    const bool fval  = (f < CF);
    const int fcl    = fval ? f : (CF - 1);
    const int c0     = fc0[fcl];
    const int c1     = fc1[fcl];

    v8f accU = {};
    v8f accV = {};
#pragma unroll
    for (int k0 = 0; k0 < DFEAT; k0 += 4) {
        v2f a;
        a.x = lf[row * DFEAT + k0 + khalf + 0];
        a.y = lf[row * DFEAT + k0 + khalf + 1];
        const int kA = k0 + khalf, kB = kA + 1;
        v2f b0, b1;
        b0.x = (fval && c0 == kA) ? 1.0f : 0.0f;
        b0.y = (fval && c0 == kB) ? 1.0f : 0.0f;
        b1.x = (fval && c1 == kA) ? 1.0f : 0.0f;
        b1.y = (fval && c1 == kB) ? 1.0f : 0.0f;
        accU = __builtin_amdgcn_wmma_f32_16x16x4_f32(
            false, a, false, b0, (short)0, accU, false, false);
        accV = __builtin_amdgcn_wmma_f32_16x16x4_f32(
            false, a, false, b1, (short)0, accV, false, false);
    }

    // C/D layout: vgpr r -> M = r + (lane>=16 ? 8 : 0), N = lane&15
    const int mbase = mtile * 16 + ((lane >> 4) << 3);
    const int col   = ntile * 16 + (lane & 15);
#pragma unroll
    for (int r = 0; r < 8; ++r) {
        U[(mbase + r) * CFPAD + col] = accU[r];
        V[(mbase + r) * CFPAD + col] = accV[r];
    }
}

// --------------------------------------------------- K3: streaming writer
// One wave handles feature-pair f and 32 consecutive sample-pairs:
// 32 rows x 24 floats = 768 floats = 6 iterations of 32 lanes x float4,
// fully coalesced (512 contiguous bytes per wave-store). 24 % 4 == 0 so each
// float4 lies in exactly one row.
__global__ void rll_writer_kernel(const char* __restrict__ ws,
                                  float* __restrict__ out) {
    const int*    fc0 = (const int*)   (ws + FC0_OFF);
    const int*    fc1 = (const int*)   (ws + FC1_OFF);
    const int2*   pq  = (const int2*)  (ws + PQ_OFF);
    const float2* rhs = (const float2*)(ws + RHS_OFF);
    const float*  U   = (const float*) (ws + U_OFF);
    const float*  V   = (const float*) (ws + V_OFF);

    const int gtid = blockIdx.x * blockDim.x + threadIdx.x;
    const int wave = gtid >> 5;
    const int lane = gtid & 31;
    const int f    = wave / SCHUNKS;
    const int s0   = (wave - f * SCHUNKS) * 32;
    const int c0   = fc0[f];
    const int c1   = fc1[f];

    const size_t base = ((size_t)f * CS + (size_t)s0) * (size_t)DFEAT;
    v4f* __restrict__ outv = (v4f*)(out + base);

#pragma unroll
    for (int it = 0; it < 6; ++it) {
        const int g  = it * 32 + lane;   // 0..191 within this wave's 768 floats
        const int rl = g / 6;            // local row (sample) 0..31
        const int ch = g - rl * 6;       // float4 chunk within row, 0..5
        const int s  = s0 + rl;

        const int2   e = pq[s];
        const float2 r = rhs[s];
        // w0 -> column c0 (sample p), w1 -> column c1 (sample q)
        const float w0 = U[e.x * CFPAD + f] * r.x + V[e.x * CFPAD + f] * r.y;
        const float w1 = U[e.y * CFPAD + f] * r.x + V[e.y * CFPAD + f] * r.y;

        v4f val = {0.0f, 0.0f, 0.0f, 0.0f};
        const int cb = ch * 4;
#pragma unroll
        for (int ee = 0; ee < 4; ++ee) {
            const int col = cb + ee;
            float x = 0.0f;
            x = (col == c0) ? w0 : x;
            x = (col == c1) ? w1 : x;
            val[ee] = x;
        }
        // output is write-once, never re-read: non-temporal store
        __builtin_nontemporal_store(val, outv + g);
    }
}

// -------------------------------------------------------------------------
extern "C" void kernel_launch(void* const* d_in, const int* in_sizes, int n_in,
                              void* d_out, int out_size, void* d_ws, size_t ws_size,
                              hipStream_t stream) {
    const float* distr = (const float*)d_in[0];   // [128, 25] f32
    char*  ws  = (char*)d_ws;
    float* out = (float*)d_out;                   // [276*8128, 24] f32

    rll_prep_kernel<<<1, 256, 0, stream>>>(distr, ws);
    rll_uv_wmma_kernel<<<18, 256, 0, stream>>>(ws);           // 144 waves
    // 70104 waves = 276 * 254; 256 threads (8 waves) per block -> 8763 blocks
    rll_writer_kernel<<<8763, 256, 0, stream>>>(ws, out);
}